// HRInexactSTCG_40561671144080
// MI455X (gfx1250) — compile-verified
//
#include <hip/hip_runtime.h>
#include <math.h>

// ---------------- scalar workspace slots ----------------
#define SC_RZ    0
#define SC_PGN   1
#define SC_PHP   2
#define SC_PP    3
#define SC_RP    4
#define SC_ALPHA 5
#define SC_SC    6
#define SC_RZN   7
#define SC_Z2    8
#define SC_CONV  9

#define MAXIT 40

typedef __attribute__((ext_vector_type(2))) float v2f;
typedef __attribute__((ext_vector_type(8))) float v8f;

static __device__ __forceinline__ v2f mk2(float2 v) {
  v2f r; r.x = v.x; r.y = v.y; return r;
}

// ---------------- WMMA GEMM: AAt += A_tile * A_tile^T ----------------
// A is M(=64) x N row-major. Each block: 512 threads = 16 waves, wave w owns
// output tile (w>>2, w&3). Blocks split the K(=N) range; partial tiles are
// atomically accumulated into AAt (zeroed beforehand). K unrolled by 16:
// 8 clustered b64 loads feed 4 WMMAs on two alternating accumulators.
__global__ void k_aat(const float* __restrict__ Amat, float* __restrict__ AAt,
                      int N, int kchunk) {
  int wave = threadIdx.x >> 5;
  int lane = threadIdx.x & 31;
  int ti = wave >> 2, tj = wave & 3;
  int half = lane >> 4, lr = lane & 15;
  long k0 = (long)blockIdx.x * kchunk;
  long k1 = k0 + kchunk; if (k1 > N) k1 = N;
  const float2* pa = (const float2*)(Amat + (size_t)(ti * 16 + lr) * N);
  const float2* pb = (const float2*)(Amat + (size_t)(tj * 16 + lr) * N);
  v8f c0 = {0.f, 0.f, 0.f, 0.f, 0.f, 0.f, 0.f, 0.f};
  v8f c1 = {0.f, 0.f, 0.f, 0.f, 0.f, 0.f, 0.f, 0.f};
  long k = k0;
  for (; k + 15 < k1; k += 16) {
    long base = (k >> 1) + half;         // float2 index; lanes 16-31 take K+2,K+3
    float2 a0 = pa[base],     b0 = pb[base];
    float2 a1 = pa[base + 2], b1 = pb[base + 2];
    float2 a2 = pa[base + 4], b2 = pb[base + 4];
    float2 a3 = pa[base + 6], b3 = pb[base + 6];
    c0 = __builtin_amdgcn_wmma_f32_16x16x4_f32(false, mk2(a0), false, mk2(b0),
                                               (short)0, c0, false, false);
    c1 = __builtin_amdgcn_wmma_f32_16x16x4_f32(false, mk2(a1), false, mk2(b1),
                                               (short)0, c1, false, false);
    c0 = __builtin_amdgcn_wmma_f32_16x16x4_f32(false, mk2(a2), false, mk2(b2),
                                               (short)0, c0, false, false);
    c1 = __builtin_amdgcn_wmma_f32_16x16x4_f32(false, mk2(a3), false, mk2(b3),
                                               (short)0, c1, false, false);
  }
  for (; k + 3 < k1; k += 4) {
    long idx = (k >> 1) + half;
    c0 = __builtin_amdgcn_wmma_f32_16x16x4_f32(false, mk2(pa[idx]), false,
                                               mk2(pb[idx]), (short)0, c0,
                                               false, false);
  }
  for (int rr = 0; rr < 8; ++rr) {
    int row = ti * 16 + rr + 8 * half;
    int col = tj * 16 + lr;
    atomicAdd(&AAt[row * 64 + col], c0[rr] + c1[rr]);
  }
}

// ---------------- zero a float region ----------------
__global__ void k_zero(float* buf, int n) {
  int i = blockIdx.x * blockDim.x + threadIdx.x;
  if (i < n) buf[i] = 0.f;
}

// ---------------- 64x64 Gauss-Jordan inverse in LDS ----------------
__global__ void k_invert(const float* __restrict__ AAt, float* __restrict__ AInv) {
  __shared__ float Mg[64 * 128];
  __shared__ float colk[64];
  int tid = threadIdx.x;                  // 128 threads
  for (int idx = tid; idx < 64 * 64; idx += 128) {
    int rr = idx >> 6, cc = idx & 63;
    Mg[rr * 128 + cc] = AAt[idx] + ((rr == cc) ? 1e-8f : 0.f);
    Mg[rr * 128 + 64 + cc] = (rr == cc) ? 1.f : 0.f;
  }
  __syncthreads();
  for (int k = 0; k < 64; ++k) {
    if (tid < 64) colk[tid] = Mg[tid * 128 + k];
    __syncthreads();
    float invp = 1.f / colk[k];
    Mg[k * 128 + tid] *= invp;
    __syncthreads();
    float pr = Mg[k * 128 + tid];
    for (int r2 = 0; r2 < 64; ++r2)
      if (r2 != k) Mg[r2 * 128 + tid] -= colk[r2] * pr;
    __syncthreads();
  }
  for (int idx = tid; idx < 64 * 64; idx += 128) {
    int rr = idx >> 6, cc = idx & 63;
    AInv[idx] = Mg[rr * 128 + 64 + cc];
  }
}

// ---------------- w[m] = dot(A row m, v), one block per row ----------------
__global__ void k_rowdot(const float* __restrict__ Amat, const float* __restrict__ v,
                         float* __restrict__ outv, int N, const float* conv) {
  if (conv && conv[0] != 0.f) return;
  int m = blockIdx.x;
  const float* row = Amat + (size_t)m * N;
  float acc = 0.f;
  for (int i = threadIdx.x; i < N; i += blockDim.x) acc += row[i] * v[i];
  for (int off = 16; off; off >>= 1) acc += __shfl_xor(acc, off, 32);
  __shared__ float red[8];
  int lane = threadIdx.x & 31, w = threadIdx.x >> 5;
  if (lane == 0) red[w] = acc;
  __syncthreads();
  if (threadIdx.x == 0) {
    float s = 0.f;
    int nw = blockDim.x >> 5;
    for (int i = 0; i < nw; ++i) s += red[i];
    outv[m] = s;
  }
}

// ---------------- y = AInv (64x64) @ w ----------------
__global__ void k_smally(const float* __restrict__ AInv, const float* __restrict__ w,
                         float* __restrict__ y, const float* conv) {
  if (conv && conv[0] != 0.f) return;
  int m = threadIdx.x;
  if (m < 64) {
    float s = 0.f;
    for (int j = 0; j < 64; ++j) s += AInv[m * 64 + j] * w[j];
    y[m] = s;
  }
}

// ---------------- r0 = -(g - A^T y); t=0; z=p=r0; rz += r0.r0 ----------------
__global__ void k_initvec(const float* __restrict__ g, const float* __restrict__ Amat,
                          const float* __restrict__ y, float* t, float* r, float* z,
                          float* p, float* scal, int N) {
  __shared__ float ys[64];
  if (threadIdx.x < 64) ys[threadIdx.x] = y[threadIdx.x];
  __syncthreads();
  int i = blockIdx.x * blockDim.x + threadIdx.x;
  float contrib = 0.f;
  if (i < N) {
    float aty = 0.f;
    for (int m = 0; m < 64; ++m) aty += Amat[(size_t)m * N + i] * ys[m];
    float r0 = -(g[i] - aty);
    t[i] = 0.f; r[i] = r0; z[i] = r0; p[i] = r0;
    contrib = r0 * r0;
  }
  for (int off = 16; off; off >>= 1) contrib += __shfl_xor(contrib, off, 32);
  if ((threadIdx.x & 31) == 0) atomicAdd(&scal[SC_RZ], contrib);
}

__global__ void k_initscal(float* scal) {
  float rz = scal[SC_RZ];
  float pgn = sqrtf(fmaxf(rz, 0.f));
  scal[SC_PGN] = pgn;
  scal[SC_CONV] = (pgn <= 1e-14f) ? 1.f : 0.f;
}

// ---------------- per-iteration prologue: clear accumulators ----------------
__global__ void k_iterpre(float* w64, float* scal) {
  if (scal[SC_CONV] != 0.f) return;
  int tid = threadIdx.x;
  if (tid < 64) w64[tid] = 0.f;
  if (tid == 0) { scal[SC_PHP] = 0.f; scal[SC_PP] = 0.f; scal[SC_RP] = 0.f; }
}

// ---------------- w64 = U^T p (coalesced: thread t -> col t&63) ----------------
__global__ void k_utp(const float* __restrict__ U, const float* __restrict__ p,
                      float* __restrict__ w64, int N, const float* scal) {
  if (scal[SC_CONV] != 0.f) return;
  int col = threadIdx.x & 63;
  int rg = threadIdx.x >> 6;                 // 0..3
  float acc = 0.f;
  for (int i = blockIdx.x * 4 + rg; i < N; i += gridDim.x * 4)
    acc += U[(size_t)i * 64 + col] * p[i];
  __shared__ float red[256];
  red[threadIdx.x] = acc;
  __syncthreads();
  if (threadIdx.x < 64) {
    float s = red[threadIdx.x] + red[threadIdx.x + 64] +
              red[threadIdx.x + 128] + red[threadIdx.x + 192];
    atomicAdd(&w64[threadIdx.x], s);
  }
}

// ---------------- Hp = diag*p + U w64; accumulate pHp, pp, rp ----------------
__global__ void k_hvp(const float* __restrict__ U, const float* __restrict__ dg,
                      const float* __restrict__ pv, const float* __restrict__ rv,
                      const float* __restrict__ w64, float* __restrict__ Hp,
                      float* scal, int N) {
  if (scal[SC_CONV] != 0.f) return;
  __shared__ float ws64[64];
  if (threadIdx.x < 64) ws64[threadIdx.x] = w64[threadIdx.x];
  __syncthreads();
  int i = blockIdx.x * blockDim.x + threadIdx.x;
  float php = 0.f, pp = 0.f, rp = 0.f;
  if (i < N) {
    const float4* urow = (const float4*)(U + (size_t)i * 64);
    __builtin_prefetch(U + (size_t)(i + 8192) * 64, 0, 0);  // speculative
    float s = 0.f;
    for (int k4 = 0; k4 < 16; ++k4) {
      float4 u = urow[k4];
      s += u.x * ws64[4 * k4] + u.y * ws64[4 * k4 + 1] +
           u.z * ws64[4 * k4 + 2] + u.w * ws64[4 * k4 + 3];
    }
    float pi = pv[i];
    float h = dg[i] * pi + s;
    Hp[i] = h;
    php = pi * h; pp = pi * pi; rp = rv[i] * pi;
  }
  for (int off = 16; off; off >>= 1) {
    php += __shfl_xor(php, off, 32);
    pp  += __shfl_xor(pp,  off, 32);
    rp  += __shfl_xor(rp,  off, 32);
  }
  if ((threadIdx.x & 31) == 0) {
    atomicAdd(&scal[SC_PHP], php);
    atomicAdd(&scal[SC_PP], pp);
    atomicAdd(&scal[SC_RP], rp);
  }
}

// ---------------- short-circuit / alpha; clear w64, hz, rzn/z2 ----------------
__global__ void k_alpha(float* scal, float* pHpD, float* w64, float* hz, int it) {
  if (scal[SC_CONV] != 0.f) return;
  int tid = threadIdx.x;
  if (tid < 64) { w64[tid] = 0.f; hz[tid] = 0.f; }
  if (tid == 0) {
    float php = scal[SC_PHP], pp = scal[SC_PP], rp = scal[SC_RP];
    float pgn = scal[SC_PGN], rz = scal[SC_RZ];
    float absf = 1e-8f * pgn * pgn;
    bool bad  = php <= fmaxf(1e-8f * pp, absf);
    bool stag = fabsf(rp) < 1e-30f;
    float sc = (bad || stag) ? 1.f : 0.f;
    scal[SC_SC] = sc;
    scal[SC_ALPHA] = (sc != 0.f) ? 0.f : rz / fmaxf(php, 1e-30f);
    pHpD[it] = php;
    scal[SC_RZN] = 0.f; scal[SC_Z2] = 0.f;
  }
}

// ---------------- t += a*p; r -= a*Hp; stash P[it], HP[it] ----------------
__global__ void k_update(float* t, float* r, const float* __restrict__ pv,
                         const float* __restrict__ Hp, float* __restrict__ Pb,
                         float* __restrict__ HPb, const float* scal, int N, int it) {
  if (scal[SC_CONV] != 0.f) return;
  float alpha = scal[SC_ALPHA];
  int i = blockIdx.x * blockDim.x + threadIdx.x;
  if (i < N) {
    float pi = pv[i], hi = Hp[i];
    t[i] += alpha * pi;
    r[i] -= alpha * hi;
    Pb[(size_t)it * N + i] = pi;
    HPb[(size_t)it * N + i] = hi;
  }
}

// ---------------- z = r - A^T y; accumulate r.z and z.z ----------------
__global__ void k_zproj(const float* __restrict__ rv, const float* __restrict__ Amat,
                        const float* __restrict__ y, float* __restrict__ z,
                        float* scal, int N) {
  if (scal[SC_CONV] != 0.f) return;
  __shared__ float ys[64];
  if (threadIdx.x < 64) ys[threadIdx.x] = y[threadIdx.x];
  __syncthreads();
  int i = blockIdx.x * blockDim.x + threadIdx.x;
  float rzp = 0.f, z2 = 0.f;
  if (i < N) {
    float aty = 0.f;
    for (int m = 0; m < 64; ++m) aty += Amat[(size_t)m * N + i] * ys[m];
    float ri = rv[i];
    float zi = ri - aty;
    z[i] = zi;
    rzp = ri * zi; z2 = zi * zi;
  }
  for (int off = 16; off; off >>= 1) {
    rzp += __shfl_xor(rzp, off, 32);
    z2  += __shfl_xor(z2,  off, 32);
  }
  if ((threadIdx.x & 31) == 0) {
    atomicAdd(&scal[SC_RZN], rzp);
    atomicAdd(&scal[SC_Z2], z2);
  }
}

// ---------------- hz[j] = dot(HP[j], z) for j<=it ----------------
__global__ void k_reorth(const float* __restrict__ HPb, const float* __restrict__ z,
                         float* __restrict__ hz, const float* scal, int N, int it) {
  if (scal[SC_CONV] != 0.f) return;
  float acc[MAXIT];
  int nj = it + 1;
  for (int j = 0; j < nj; ++j) acc[j] = 0.f;
  for (int i = blockIdx.x * blockDim.x + threadIdx.x; i < N;
       i += gridDim.x * blockDim.x) {
    float zi = z[i];
    for (int j = 0; j < nj; ++j) acc[j] += HPb[(size_t)j * N + i] * zi;
  }
  for (int j = 0; j < nj; ++j) {
    float a = acc[j];
    for (int off = 16; off; off >>= 1) a += __shfl_xor(a, off, 32);
    if ((threadIdx.x & 31) == 0) atomicAdd(&hz[j], a);
  }
}

__global__ void k_coeffs(const float* hz, const float* pHpD, float* cf,
                         const float* scal, int it) {
  if (scal[SC_CONV] != 0.f) return;
  int j = threadIdx.x;
  if (j < 64) {
    float d = pHpD[j];
    float ds = (fabsf(d) > 1e-30f) ? d : 1e-30f;
    cf[j] = (j <= it) ? hz[j] / ds : 0.f;
  }
}

// ---------------- p = z - sum_j cf[j] * P[j] ----------------
__global__ void k_pnew(const float* __restrict__ z, const float* __restrict__ Pb,
                       const float* __restrict__ cf, float* __restrict__ pv,
                       const float* scal, int N, int it) {
  if (scal[SC_CONV] != 0.f) return;
  __shared__ float cs[MAXIT];
  if ((int)threadIdx.x <= it) cs[threadIdx.x] = cf[threadIdx.x];
  __syncthreads();
  int i = blockIdx.x * blockDim.x + threadIdx.x;
  if (i < N) {
    float s = 0.f;
    for (int j = 0; j <= it; ++j) s += cs[j] * Pb[(size_t)j * N + i];
    pv[i] = z[i] - s;
  }
}

__global__ void k_endstep(float* scal) {
  if (scal[SC_CONV] != 0.f) return;
  float zn = sqrtf(fmaxf(scal[SC_Z2], 0.f));
  float pgn = scal[SC_PGN];
  float tol = fmaxf(1e-14f, 1e-6f * pgn);
  scal[SC_RZ] = scal[SC_RZN];
  if (zn <= tol || scal[SC_SC] != 0.f) scal[SC_CONV] = 1.f;
}

__global__ void k_out(const float* __restrict__ t, const float* scal,
                      float* __restrict__ out, int N) {
  int i = blockIdx.x * blockDim.x + threadIdx.x;
  if (i < N) out[i] = t[i];
  if (i == 0) {
    out[N] = scal[SC_PGN];
    out[N + 1] = (scal[SC_CONV] != 0.f) ? 1.f : 0.f;
  }
}

// =====================================================================
extern "C" void kernel_launch(void* const* d_in, const int* in_sizes, int n_in,
                              void* d_out, int out_size, void* d_ws, size_t ws_size,
                              hipStream_t stream) {
  (void)n_in; (void)out_size; (void)ws_size;
  const float* g    = (const float*)d_in[0];
  const float* dg   = (const float*)d_in[1];
  const float* U    = (const float*)d_in[2];
  const float* Amat = (const float*)d_in[3];
  int N = in_sizes[0];

  // -------- workspace layout (floats) --------
  float* ws = (float*)d_ws;
  size_t o = 0;
  float* t    = ws + o; o += (size_t)N;
  float* r    = ws + o; o += (size_t)N;
  float* z    = ws + o; o += (size_t)N;
  float* p    = ws + o; o += (size_t)N;
  float* Hp   = ws + o; o += (size_t)N;
  float* Pb   = ws + o; o += (size_t)MAXIT * N;
  float* HPb  = ws + o; o += (size_t)MAXIT * N;
  float* AAt  = ws + o; o += 64 * 64;
  float* AInv = ws + o; o += 64 * 64;
  float* w64  = ws + o; o += 64;
  float* y64  = ws + o; o += 64;
  float* pHpD = ws + o; o += 64;
  float* hz   = ws + o; o += 64;
  float* cf   = ws + o; o += 64;
  float* scal = ws + o; o += 64;
  const float* conv = scal + SC_CONV;

  const int NB = (N + 255) / 256;

  // ---- zero AAt..scal region (contiguous) ----
  int zcnt = 64 * 64 * 2 + 64 * 6;
  k_zero<<<(zcnt + 255) / 256, 256, 0, stream>>>(AAt, zcnt);

  // ---- AAt = A A^T via WMMA f32 16x16x4 (K unrolled x4, dual accumulators) ----
  const int AAT_BLOCKS = 25;
  int kchunk = ((N + AAT_BLOCKS - 1) / AAT_BLOCKS + 3) & ~3;
  k_aat<<<AAT_BLOCKS, 512, 0, stream>>>(Amat, AAt, N, kchunk);

  // ---- (AAt + 1e-8 I)^-1 ----
  k_invert<<<1, 128, 0, stream>>>(AAt, AInv);

  // ---- r0 = -project(g); init t,z,p, rz, pgn, conv ----
  k_rowdot<<<64, 256, 0, stream>>>(Amat, g, w64, N, nullptr);
  k_smally<<<1, 64, 0, stream>>>(AInv, w64, y64, nullptr);
  k_initvec<<<NB, 256, 0, stream>>>(g, Amat, y64, t, r, z, p, scal, N);
  k_initscal<<<1, 1, 0, stream>>>(scal);

  // ---- 40 CG iterations (conv flag gates all work) ----
  for (int it = 0; it < MAXIT; ++it) {
    k_iterpre<<<1, 64, 0, stream>>>(w64, scal);
    k_utp<<<128, 256, 0, stream>>>(U, p, w64, N, scal);
    k_hvp<<<NB, 256, 0, stream>>>(U, dg, p, r, w64, Hp, scal, N);
    k_alpha<<<1, 64, 0, stream>>>(scal, pHpD, w64, hz, it);
    k_update<<<NB, 256, 0, stream>>>(t, r, p, Hp, Pb, HPb, scal, N, it);
    k_rowdot<<<64, 256, 0, stream>>>(Amat, r, w64, N, conv);
    k_smally<<<1, 64, 0, stream>>>(AInv, w64, y64, conv);
    k_zproj<<<NB, 256, 0, stream>>>(r, Amat, y64, z, scal, N);
    k_reorth<<<104, 256, 0, stream>>>(HPb, z, hz, scal, N, it);
    k_coeffs<<<1, 64, 0, stream>>>(hz, pHpD, cf, scal, it);
    k_pnew<<<NB, 256, 0, stream>>>(z, Pb, cf, p, scal, N, it);
    k_endstep<<<1, 1, 0, stream>>>(scal);
  }

  // ---- outputs: t (N), proj_grad_norm, converged ----
  k_out<<<NB, 256, 0, stream>>>(t, scal, (float*)d_out, N);
}